// SEE_15857019257345
// MI455X (gfx1250) — compile-verified
//
#include <hip/hip_runtime.h>

// MoE expert-MLP kernel for gfx1250 (MI455X), fp32 WMMA path, round 2.
// B=8, N=4096, T=96, K=4, E=8, H=256  -> BN = 32768 tokens.
//
// Per token: out = route_scale * (x + sum_e c_{n,e} * (relu(x W1_e + b1_e) W2_e + b2_e))
// with c_{n,e} = sum_k normw_k * mask_k * [idx_k == e].  util[n][e] = any active route.

typedef float v2f __attribute__((ext_vector_type(2)));
typedef float v8f __attribute__((ext_vector_type(8)));

#define T_DIM 96
#define H_DIM 256
#define E_NUM 8
#define K_RT  4
#define BN_TOK 32768

// LDS strides in floats. Bank math (64 banks, fp32 words):
//  W1TS=116: 116 mod 64 = 52, gcd(52,64)=4; half-wave +2 offset not divisible by 4
//            -> all 32 lanes' b64 loads conflict-free.
//  W2TS=260: 260 mod 64 = 4, same gcd-4 / +2 argument -> conflict-free.
//  XS=100, HS=20: 36*l16 / 20*l16 all distinct mod 64 -> conflict-free b64.
#define W1TS 116   // W1 transposed: 256 rows (h) x 96 cols (t)
#define W2TS 260   // W2 transposed:  96 rows (t) x 256 cols (h)
#define XS   100   // per-wave X tile: 16 x 96
#define HS   20    // per-wave h-chunk scratch: 16 x 16

// float offsets inside dynamic LDS
#define OFF_W1T 0
#define OFF_W2T (OFF_W1T + 256 * W1TS)        // 29696
#define OFF_B1  (OFF_W2T + 96 * W2TS)         // 54656
#define OFF_B2  (OFF_B1 + 256)                // 54912
#define OFF_C   (OFF_B2 + 96)                 // 55008 (cArr: 8 waves x 16)
#define OFF_X   (OFF_C + 128)                 // 55136
#define OFF_H   (OFF_X + 8 * 16 * XS)         // 67936
#define SMEM_FLOATS (OFF_H + 8 * 16 * HS)     // 70496 floats = 281984 B < 320 KB

__global__ __launch_bounds__(256, 1)
void moe_wmma_f32_kernel(const float* __restrict__ x,
                         const int* __restrict__ cidx,
                         const float* __restrict__ cw,
                         const unsigned char* __restrict__ cmask,
                         const float* __restrict__ route_scale,
                         const float* __restrict__ W1,
                         const float* __restrict__ b1,
                         const float* __restrict__ W2,
                         const float* __restrict__ b2,
                         float* __restrict__ out_enh,
                         float* __restrict__ out_util)
{
    extern __shared__ float smem[];
    float* w1t  = smem + OFF_W1T;
    float* w2t  = smem + OFF_W2T;
    float* b1s  = smem + OFF_B1;
    float* b2s  = smem + OFF_B2;
    float* cArr = smem + OFF_C;

    const int tid  = threadIdx.x;
    const int wave = tid >> 5;
    const int lane = tid & 31;
    const int half = lane >> 4;   // 0: lanes 0-15, 1: lanes 16-31
    const int l16  = lane & 15;

    float* xs = smem + OFF_X + wave * 16 * XS;
    float* hs = smem + OFF_H + wave * 16 * HS;

    const int tileBase = blockIdx.x * 128 + wave * 16;   // 16 tokens per wave

    // ---- stage this wave's X tile [16 x 96] into LDS ----
    {
        const float4* xg = (const float4*)(x + (size_t)tileBase * T_DIM);
        for (int i = lane; i < (16 * T_DIM) / 4; i += 32) {
            float4 v = xg[i];
            int fi = i * 4;
            int r = fi / T_DIM, c0 = fi % T_DIM;
            float* d = xs + r * XS + c0;
            d[0] = v.x; d[1] = v.y; d[2] = v.z; d[3] = v.w;
        }
    }
    asm volatile("" ::: "memory");   // wave-private LDS; HW keeps per-wave DS in order

    // ---- preload all 24 GEMM1 A-fragments (X is expert/chunk invariant) ----
    // A layout 16x4 f32: M = l16; VGPR0/1 hold K = {0,1} (lanes 0-15) / {2,3} (16-31)
    v2f aX[24];
    {
        const float* arow = xs + l16 * XS + half * 2;
        #pragma unroll
        for (int kk = 0; kk < 24; ++kk)
            aX[kk] = *(const v2f*)(arow + kk * 4);
    }

    // ---- per-lane routing data for token row r = l16 (both halves redundant) ----
    const int tok = tileBase + l16;
    int   ridx[K_RT];
    float nw[K_RT];
    int   rm[K_RT];
    {
        float rw[K_RT];
        float wsum = 0.f;
        #pragma unroll
        for (int k = 0; k < K_RT; ++k) {
            ridx[k] = cidx[tok * K_RT + k];
            rw[k]   = cw[tok * K_RT + k];
            rm[k]   = (int)cmask[tok * K_RT + k];
            wsum   += rw[k];
        }
        const float inv = 1.f / fmaxf(wsum, 1e-12f);
        #pragma unroll
        for (int k = 0; k < K_RT; ++k) nw[k] = rw[k] * inv;
    }

    // ---- utilization output (lanes 0-15 only) ----
    if (lane < 16) {
        #pragma unroll
        for (int e = 0; e < E_NUM; ++e) {
            float u = 0.f;
            #pragma unroll
            for (int k = 0; k < K_RT; ++k)
                if (rm[k] && ridx[k] == e) u = 1.f;
            out_util[(size_t)tok * E_NUM + e] = u;
        }
    }

    // ---- kernel-invariant per-lane GEMM2 B base pointers (imm-offset friendly) ----
    // B element = W2t[t = nt*16 + l16][h], pairs (h0,h0+1) contiguous.
    const float* pB0 = w2t + l16 * W2TS + half * 2;            // nt = 0..3
    const float* pB1 = pB0 + 4 * 16 * W2TS;                    // nt = 4..5

    // ---- persistent accumulators: 16 x 96 output tile in WMMA C layout ----
    v8f acc[6];
    const v8f vzero = {0.f, 0.f, 0.f, 0.f, 0.f, 0.f, 0.f, 0.f};
    #pragma unroll
    for (int nt = 0; nt < 6; ++nt) acc[nt] = vzero;

    for (int e = 0; e < E_NUM; ++e) {
        __syncthreads();  // previous expert's compute done before LDS overwrite
        // ---- cooperative stage of W1_e^T, W2_e^T, b1_e, b2_e into LDS ----
        {
            const float4* w1g = (const float4*)(W1 + (size_t)e * 96 * 256);
            for (int i = tid; i < (96 * 256) / 4; i += 256) {
                float4 v = w1g[i];
                int fi = i * 4;
                int t = fi >> 8, h0 = fi & 255;       // global W1 is [t][h]
                w1t[(h0 + 0) * W1TS + t] = v.x;
                w1t[(h0 + 1) * W1TS + t] = v.y;
                w1t[(h0 + 2) * W1TS + t] = v.z;
                w1t[(h0 + 3) * W1TS + t] = v.w;
            }
            const float4* w2g = (const float4*)(W2 + (size_t)e * 256 * 96);
            for (int i = tid; i < (256 * 96) / 4; i += 256) {
                float4 v = w2g[i];
                int fi = i * 4;
                int h = fi / 96, t0 = fi % 96;        // global W2 is [h][t]
                w2t[(t0 + 0) * W2TS + h] = v.x;
                w2t[(t0 + 1) * W2TS + h] = v.y;
                w2t[(t0 + 2) * W2TS + h] = v.z;
                w2t[(t0 + 3) * W2TS + h] = v.w;
            }
            b1s[tid] = b1[e * 256 + tid];
            if (tid < 96) b2s[tid] = b2[e * 96 + tid];
        }
        __syncthreads();

        // ---- prefetch next expert's weights into cache while we compute ----
        if (e + 1 < E_NUM) {
            const char* n1 = (const char*)(W1 + (size_t)(e + 1) * 96 * 256);
            const char* n2 = (const char*)(W2 + (size_t)(e + 1) * 256 * 96);
            #pragma unroll
            for (int j = 0; j < 3; ++j) {
                __builtin_prefetch(n1 + tid * 384 + j * 128, 0, 1);
                __builtin_prefetch(n2 + tid * 384 + j * 128, 0, 1);
            }
        }

        // ---- mixing coefficient for this lane's token row ----
        float c = 0.f;
        #pragma unroll
        for (int k = 0; k < K_RT; ++k)
            if (rm[k] && ridx[k] == e) c += nw[k];

        if (lane < 16) cArr[wave * 16 + l16] = c;
        asm volatile("" ::: "memory");

        // ---- fold c * b2_e into accumulators (C layout: col = nt*16+l16) ----
        #pragma unroll
        for (int nt = 0; nt < 6; ++nt) {
            float bv = b2s[nt * 16 + l16];
            #pragma unroll
            for (int v = 0; v < 8; ++v)
                acc[nt][v] += cArr[wave * 16 + v + half * 8] * bv;
        }

        // ---- fused GEMM1 (16x96 @ 96x256) + GEMM2 (16x256 @ 256x96), H-chunks of 16 ----
        const float* bcol = w1t + l16 * W1TS + half * 2;   // + ch*16*W1TS per chunk
        const float* q0 = pB0;                             // + ch*16 per chunk
        const float* q1 = pB1;
        for (int ch = 0; ch < 16; ++ch) {
            // GEMM1: h_chunk[16x16] = X @ W1[:, ch*16 : ch*16+16]
            v8f h = vzero;
            #pragma unroll
            for (int kk = 0; kk < 24; ++kk) {
                v2f b = *(const v2f*)(bcol + kk * 4);      // {W1[k0][col], W1[k0+1][col]}
                h = __builtin_amdgcn_wmma_f32_16x16x4_f32(
                        false, aX[kk], false, b, (short)0, h, false, false);
            }
            // bias + relu (bias uniform over rows in C layout: col = ch*16+l16)
            const float bias = b1s[ch * 16 + l16];
            #pragma unroll
            for (int v = 0; v < 8; ++v) h[v] = fmaxf(h[v] + bias, 0.f);

            // C-layout -> A-layout via per-wave LDS scratch
            {
                float* drow = hs + (half * 8) * HS + l16;
                #pragma unroll
                for (int v = 0; v < 8; ++v) drow[v * HS] = h[v];
            }
            asm volatile("" ::: "memory");  // per-wave DS is HW in-order; block reorder

            // GEMM2: acc += (c .* h_chunk) @ W2[ch*16 : ch*16+16, :]
            const float* a2row = hs + l16 * HS + half * 2;
            #pragma unroll
            for (int kk2 = 0; kk2 < 4; ++kk2) {
                v2f a2 = *(const v2f*)(a2row + kk2 * 4);
                a2[0] *= c; a2[1] *= c;
                #pragma unroll
                for (int nt = 0; nt < 6; ++nt) {
                    const float* bp = (nt < 4) ? (q0 + nt * (16 * W2TS) + kk2 * 4)
                                               : (q1 + (nt - 4) * (16 * W2TS) + kk2 * 4);
                    v2f bb = *(const v2f*)bp;              // {W2[h0][col], W2[h0+1][col]}
                    acc[nt] = __builtin_amdgcn_wmma_f32_16x16x4_f32(
                                  false, a2, false, bb, (short)0, acc[nt], false, false);
                }
            }
            asm volatile("" ::: "memory");  // scratch reads before next chunk's stores

            bcol += 16 * W1TS;   // next 16 H-columns of W1
            q0   += 16;          // next 16 H-rows of W2 (h0 += 16)
            q1   += 16;
        }
    }

    // ---- epilogue: out = route_scale * (x + acc); 64B-segment coalesced stores ----
    const float rs = route_scale[0];
    #pragma unroll
    for (int nt = 0; nt < 6; ++nt) {
        #pragma unroll
        for (int v = 0; v < 8; ++v) {
            const int row = v + half * 8;
            const int t   = nt * 16 + l16;
            const float val = rs * (xs[row * XS + t] + acc[nt][v]);
            out_enh[(size_t)(tileBase + row) * T_DIM + t] = val;
        }
    }
}

extern "C" void kernel_launch(void* const* d_in, const int* in_sizes, int n_in,
                              void* d_out, int out_size, void* d_ws, size_t ws_size,
                              hipStream_t stream) {
    (void)in_sizes; (void)n_in; (void)out_size; (void)d_ws; (void)ws_size;
    const float*         x    = (const float*)d_in[0];
    const int*           ci   = (const int*)d_in[1];
    const float*         cw   = (const float*)d_in[2];
    const unsigned char* cm   = (const unsigned char*)d_in[3];  // jnp.bool_ = 1 byte
    const float*         rsc  = (const float*)d_in[4];
    const float*         W1   = (const float*)d_in[5];
    const float*         b1   = (const float*)d_in[6];
    const float*         W2   = (const float*)d_in[7];
    const float*         b2   = (const float*)d_in[8];

    float* enh  = (float*)d_out;
    float* util = enh + (size_t)BN_TOK * T_DIM;

    const dim3 grid(BN_TOK / 128);   // 256 blocks, 8 waves x 16 tokens each
    const dim3 block(256);
    const size_t smem = (size_t)SMEM_FLOATS * sizeof(float);   // ~275 KB < 320 KB/WGP
    moe_wmma_f32_kernel<<<grid, block, smem, stream>>>(
        x, ci, cw, cm, rsc, W1, b1, W2, b2, enh, util);
}